// AttentionPool_10153302687757
// MI455X (gfx1250) — compile-verified
//
#include <hip/hip_runtime.h>

#define D    1024
#define BSZ  8
#define TLEN 4096
#define G    64
#define BT   (BSZ * TLEN)                 // 32768 tokens
#define EPS  1.1920928955078125e-7f       // finfo(float32).eps
#define SCALE 0.03125f                    // 1/sqrt(1024)

typedef float v2f __attribute__((ext_vector_type(2)));
typedef float v8f __attribute__((ext_vector_type(8)));

// k0: qn[d] = norm_w[d] * sum_e q[e] * Wk[e*D + d]   (grid 4 x 256)
__global__ void k0_qn(const float* __restrict__ q, const float* __restrict__ nw,
                      const float* __restrict__ Wk, float* __restrict__ qn) {
    int d = blockIdx.x * 256 + threadIdx.x;
    float acc = 0.f;
    for (int e = 0; e < D; ++e) acc += q[e] * Wk[(size_t)e * D + d];
    qn[d] = acc * nw[d];
}

// k1: per-token rms + raw score, one wave per token (grid 4096 x 256)
__global__ void k1_score(const float* __restrict__ x, const float* __restrict__ qn,
                         float* __restrict__ rms, float* __restrict__ score) {
    int wid = threadIdx.x >> 5, lane = threadIdx.x & 31;
    int tok = blockIdx.x * 8 + wid;
    const float4* xp = (const float4*)(x + (size_t)tok * D);
    const float4* qp = (const float4*)qn;
    float ss = 0.f, sc = 0.f;
#pragma unroll
    for (int j = 0; j < 8; ++j) {
        float4 xv = xp[lane + 32 * j];
        float4 qv = qp[lane + 32 * j];
        ss += xv.x * xv.x + xv.y * xv.y + xv.z * xv.z + xv.w * xv.w;
        sc += xv.x * qv.x + xv.y * qv.y + xv.z * qv.z + xv.w * qv.w;
    }
    for (int off = 16; off; off >>= 1) {
        ss += __shfl_xor(ss, off);
        sc += __shfl_xor(sc, off);
    }
    if (lane == 0) {
        float r = rsqrtf(ss * (1.0f / D) + EPS);
        rms[tok]   = r;
        score[tok] = sc * r * SCALE;
    }
}

// k2: segment max + sum(exp) per (b,g)  (grid 512 x 256)
__global__ void k2_seg(const float* __restrict__ score, const int* __restrict__ gid,
                       float* __restrict__ gmax, float* __restrict__ ginv) {
    __shared__ float red[256];
    int b = blockIdx.x >> 6, g = blockIdx.x & 63;
    const float* s  = score + b * TLEN;
    const int*   gd = gid   + b * TLEN;
    float m = -__builtin_inff();
    for (int t = threadIdx.x; t < TLEN; t += 256)
        if (gd[t] == g) m = fmaxf(m, s[t]);
    red[threadIdx.x] = m; __syncthreads();
    for (int o = 128; o; o >>= 1) {
        if (threadIdx.x < o) red[threadIdx.x] = fmaxf(red[threadIdx.x], red[threadIdx.x + o]);
        __syncthreads();
    }
    m = red[0];
    __syncthreads();
    bool empty = !(m > -__builtin_inff());
    float mm = empty ? 0.f : m;
    float acc = 0.f;
    for (int t = threadIdx.x; t < TLEN; t += 256)
        if (gd[t] == g) acc += expf(s[t] - mm);
    red[threadIdx.x] = acc; __syncthreads();
    for (int o = 128; o; o >>= 1) {
        if (threadIdx.x < o) red[threadIdx.x] += red[threadIdx.x + o];
        __syncthreads();
    }
    if (threadIdx.x == 0) {
        gmax[blockIdx.x] = mm;
        ginv[blockIdx.x] = empty ? 0.f : 1.0f / red[0];
    }
}

// k2b: per-token combined weight (softmax weight * rms)  (grid 128 x 256)
__global__ void k2b_w(const float* __restrict__ score, const int* __restrict__ gid,
                      const float* __restrict__ gmax, const float* __restrict__ ginv,
                      const float* __restrict__ rms, float* __restrict__ wtok) {
    int t  = blockIdx.x * 256 + threadIdx.x;
    int bg = (t >> 12) * G + gid[t];
    wtok[t] = expf(score[t] - gmax[bg]) * ginv[bg] * rms[t];
}

// k3: pooled partials. block = (chunk, b, d-slice of 128); LDS acc[g][128],
// each thread owns its d-column -> no atomics/barriers. (grid 256 x 128)
__global__ __launch_bounds__(128) void k3_pool(const float* __restrict__ x,
        const int* __restrict__ gid, const float* __restrict__ wtok,
        float* __restrict__ part) {
    __shared__ float acc[G * 128];
    int bx = blockIdx.x;
    int chunk = bx & 3, b = (bx >> 2) & 7, ds = bx >> 5;
    int tid = threadIdx.x;
    int d = ds * 128 + tid;
    for (int g = 0; g < G; ++g) acc[g * 128 + tid] = 0.f;
    int tbase = b * TLEN + chunk * 1024;
    for (int i = 0; i < 1024; ++i) {
        int tok = tbase + i;
        float wv = wtok[tok];
        int g = gid[tok];
        acc[g * 128 + tid] += wv * x[(size_t)tok * D + d];
    }
    float* pp = part + (size_t)chunk * (BSZ * G * D) + (size_t)(b * G) * D + d;
    for (int g = 0; g < G; ++g) pp[(size_t)g * D] = acc[g * 128 + tid];
}

// k4: reduce 4 chunk partials, fold norm_w  (grid 2048 x 256)
__global__ void k4_reduce(const float* __restrict__ part, const float* __restrict__ nw,
                          float* __restrict__ pooledW) {
    int idx = blockIdx.x * 256 + threadIdx.x;   // 0 .. B*G*D-1
    int d = idx & (D - 1);
    float s = 0.f;
#pragma unroll
    for (int c = 0; c < 4; ++c) s += part[(size_t)c * (BSZ * G * D) + idx];
    pooledW[idx] = s * nw[d];
}

// k5: out[bg, e] = sum_d pooledW[bg, d] * Wv[e, d]  via V_WMMA_F32_16X16X4_F32
// One wave per 16x16 tile. A frag: lanes0-15 rows M, K split 2*hi. B[k][n]=Wv[n][k].
__global__ __launch_bounds__(256) void k5_gemm(const float* __restrict__ A,
        const float* __restrict__ Wv, float* __restrict__ out) {
    int lane = threadIdx.x & 31, wid = threadIdx.x >> 5;
    int tile = blockIdx.x * 8 + wid;            // 0..2047
    int tm = tile >> 6, tn = tile & 63;         // 32 x 64 tiles
    int hi = lane >> 4, l15 = lane & 15;
    const float* ap = A  + (size_t)(tm * 16 + l15) * D + 2 * hi;
    const float* bp = Wv + (size_t)(tn * 16 + l15) * D + 2 * hi;
    v8f c = {};
#pragma unroll 8
    for (int k = 0; k < D; k += 4) {
        v2f a = *(const v2f*)(ap + k);
        v2f b = *(const v2f*)(bp + k);
        c = __builtin_amdgcn_wmma_f32_16x16x4_f32(false, a, false, b,
                                                  (short)0, c, false, false);
    }
    float* op = out + (size_t)(tm * 16 + 8 * hi) * D + tn * 16 + l15;
#pragma unroll
    for (int r = 0; r < 8; ++r) op[(size_t)r * D] = c[r];
}

extern "C" void kernel_launch(void* const* d_in, const int* in_sizes, int n_in,
                              void* d_out, int out_size, void* d_ws, size_t ws_size,
                              hipStream_t stream) {
    const float* x   = (const float*)d_in[0];
    const int*   gid = (const int*)  d_in[1];
    // d_in[2] = num_groups (fixed at 64)
    const float* q   = (const float*)d_in[3];
    const float* nw  = (const float*)d_in[4];
    const float* Wk  = (const float*)d_in[5];
    const float* Wv  = (const float*)d_in[6];
    float* out = (float*)d_out;

    float* ws      = (float*)d_ws;
    float* qn      = ws;                       // D
    float* rms     = qn    + D;                // BT
    float* score   = rms   + BT;               // BT
    float* gmax    = score + BT;               // B*G
    float* ginv    = gmax  + BSZ * G;          // B*G
    float* wtok    = ginv  + BSZ * G;          // BT
    float* part    = wtok  + BT;               // 4 * B*G*D
    float* pooledW = part  + (size_t)4 * BSZ * G * D;  // B*G*D
    // total ws usage ~10.9 MB

    k0_qn    <<<4,    256, 0, stream>>>(q, nw, Wk, qn);
    k1_score <<<4096, 256, 0, stream>>>(x, qn, rms, score);
    k2_seg   <<<512,  256, 0, stream>>>(score, gid, gmax, ginv);
    k2b_w    <<<128,  256, 0, stream>>>(score, gid, gmax, ginv, rms, wtok);
    k3_pool  <<<256,  128, 0, stream>>>(x, gid, wtok, part);
    k4_reduce<<<2048, 256, 0, stream>>>(part, nw, pooledW);
    k5_gemm  <<<256,  256, 0, stream>>>(pooledW, Wv, out);
}